// LatLonInterpolation_11656541241652
// MI455X (gfx1250) — compile-verified
//
#include <hip/hip_runtime.h>

// Problem constants (from the reference)
#define H_IN   721
#define W_IN   1440
#define H_OUT  720
#define W_OUT  1440
#define BC     64                    // B*C = 2*32 planes
#define PLANE_IN  (H_IN * W_IN)      // 1,038,240
#define PLANE_OUT (H_OUT * W_OUT)    // 1,036,800
#define NPIX      (H_OUT * W_OUT)    // 1,036,800 (divisible by 256 -> 4050 blocks)
#define TPB       256

typedef unsigned int u32x4 __attribute__((ext_vector_type(4)));
typedef int          i32x4 __attribute__((ext_vector_type(4)));
typedef int          i32x8 __attribute__((ext_vector_type(8)));
typedef __attribute__((address_space(3))) void lds_void;

#define HAS_TDM __has_builtin(__builtin_amdgcn_tensor_load_to_lds)

// Issue one 1-D TDM tensor load: nelem fp32 elements from gsrc -> LDS byte
// address lds_byte_addr. Descriptor per CDNA5 ISA:
//   group0: count=1 | lds_addr | global_addr[56:0] | type=2
//   group1: data_size=4B, tensor_dim0=tile_dim0=nelem, tensor_dim1=1
__device__ __forceinline__ void tdm_load_1d(unsigned lds_byte_addr,
                                            const float* gsrc,
                                            unsigned nelem) {
#if HAS_TDM
  unsigned long long ga = (unsigned long long)gsrc;
  u32x4 g0 = { 1u,                                   // count=1, user mode
               lds_byte_addr,                        // LDS dest (bytes)
               (unsigned)(ga & 0xFFFFFFFFu),         // global_addr[31:0]
               ((unsigned)((ga >> 32) & 0x01FFFFFFu))// global_addr[56:32]
                 | 0x80000000u };                    // type=2 ("image")
  i32x8 g1 = { (int)0x00020000u,        // data_size=2 (4 bytes), wg_mask=0
               (int)(nelem << 16),      // tensor_dim0[15:0]
               (int)0x00010000u,        // tensor_dim1 = 1
               (int)(nelem << 16),      // tile_dim0 = nelem
               0,                       // tile_dim1=0 (1-D), tile_dim2=0
               (int)nelem,              // tensor_dim0_stride (unused for 1-D)
               0, 0 };
  i32x4 z4 = { 0, 0, 0, 0 };
#if defined(__clang_major__) && (__clang_major__ >= 23)
  i32x8 z8 = { 0, 0, 0, 0, 0, 0, 0, 0 };
  __builtin_amdgcn_tensor_load_to_lds(g0, g1, z4, z4, z8, 0);
#else
  __builtin_amdgcn_tensor_load_to_lds(g0, g1, z4, z4, 0);
#endif
#else
  (void)lds_byte_addr; (void)gsrc; (void)nelem;
#endif
}

__global__ __launch_bounds__(TPB) void latlon_bilinear_kernel(
    const float* __restrict__ values,
    const float* __restrict__ i_map,
    const float* __restrict__ j_map,
    float* __restrict__ out)
{
  __shared__ float s_maps[2 * TPB];   // [0..255]=i tile, [256..511]=j tile
  const int tid = threadIdx.x;
  const int p   = blockIdx.x * TPB + tid;   // spatial point, grid covers NPIX exactly

#if HAS_TDM
  // Wave 0 stages the block's map tiles into LDS via the Tensor Data Mover.
  if (tid < 32) {
    unsigned lds_base = (unsigned)(unsigned long long)(lds_void*)s_maps;
    tdm_load_1d(lds_base,              i_map + (size_t)blockIdx.x * TPB, TPB);
    tdm_load_1d(lds_base + TPB * 4u,   j_map + (size_t)blockIdx.x * TPB, TPB);
#if __has_builtin(__builtin_amdgcn_s_wait_tensorcnt)
    __builtin_amdgcn_s_wait_tensorcnt(0);
#else
    asm volatile("s_wait_tensorcnt 0x0" ::: "memory");
#endif
  }
  __syncthreads();
  const float fi = s_maps[tid];
  const float fj = s_maps[TPB + tid];
#else
  const float fi = i_map[p];
  const float fj = j_map[p];
#endif

  // Maps are pre-clipped to [0, H_IN-2] / [0, W_IN-2] and non-negative,
  // so integer truncation == floor and i0+1/j0+1 are in range.
  const int   i0 = (int)fi;
  const int   j0 = (int)fj;
  const float di = fi - (float)i0;
  const float dj = fj - (float)j0;

  const float* src = values + ((size_t)i0 * W_IN + j0);
  float*       dst = out + p;

#pragma unroll 4
  for (int c = 0; c < BC; ++c) {
    const float* b = src + (size_t)c * PLANE_IN;
    // Prefetch next plane's two gather cachelines (speculative; OOB at the
    // last plane is silently dropped by the prefetch path).
    __builtin_prefetch(b + PLANE_IN, 0, 3);
    __builtin_prefetch(b + PLANE_IN + W_IN, 0, 3);
    const float f00 = b[0];
    const float f01 = b[1];
    const float f10 = b[W_IN];
    const float f11 = b[W_IN + 1];
    const float f0 = __builtin_fmaf(dj, f01 - f00, f00);   // lerp along j
    const float f1 = __builtin_fmaf(dj, f11 - f10, f10);
    const float r  = __builtin_fmaf(di, f1 - f0, f0);      // lerp along i
    // Output is write-once / never re-read: non-temporal store (TH=NT) so the
    // 265 MB output stream does not evict the L2-resident gather planes.
    __builtin_nontemporal_store(r, dst + (size_t)c * PLANE_OUT);
  }
}

extern "C" void kernel_launch(void* const* d_in, const int* in_sizes, int n_in,
                              void* d_out, int out_size, void* d_ws, size_t ws_size,
                              hipStream_t stream) {
  (void)in_sizes; (void)n_in; (void)out_size; (void)d_ws; (void)ws_size;
  const float* values = (const float*)d_in[0];
  const float* i_map  = (const float*)d_in[1];
  const float* j_map  = (const float*)d_in[2];
  float* out = (float*)d_out;

  dim3 grid(NPIX / TPB);   // 4050 blocks, 256 threads = 8 wave32 each
  hipLaunchKernelGGL(latlon_bilinear_kernel, grid, dim3(TPB), 0, stream,
                     values, i_map, j_map, out);
}